// CrfRnn_65326452572550
// MI455X (gfx1250) — compile-verified
//
#include <hip/hip_runtime.h>

// CRF-RNN mean-field on MI455X. Memory-bound: two 420MB f32 matrix streams per
// iteration (23.3 TB/s HBM -> ~180us floor). Keep f32, V_WMMA_F32_16X16X4_F32.
// One wave per 16-row M-strip computes the A fragment once and feeds 3 WMMAs
// (full N=48); K is split across waves into partial buffers for occupancy.
// exp() uses the raw v_exp_f32 intrinsic: arguments are <= 0, so the libm
// denormal-range fixup (cmp/cndmask/ldexp per exp) is dead weight.

typedef __attribute__((ext_vector_type(2))) float v2f;
typedef __attribute__((ext_vector_type(8))) float v8f;

#define L   42
#define LP  48                      // padded label dim (3 x 16 WMMA N-tiles)
#define NUM_ITER 5
// exp(-d2/(2*gamma^2)) with gamma=0.5 => exp(-2*d2) = 2^(d2 * -2*log2(e))
#define NEG2LOG2E (-2.885390081777927f)
#define LOG2E     (1.4426950408889634f)

#define WMMA_F32(A, B, C) \
    __builtin_amdgcn_wmma_f32_16x16x4_f32(false, (A), false, (B), (short)0, (C), false, false)

// Raw v_exp_f32 (no underflow fixup). Safe here: x <= 0 always.
__device__ __forceinline__ float fast_exp2(float x) { return __builtin_amdgcn_exp2f(x); }

// ---------------------------------------------------------------------------
// U = log(clip(def_idxs,1e-6,1)) ; q = softmax(U) ; zero pad columns.
// ---------------------------------------------------------------------------
__global__ void init_u_q(const float* __restrict__ defIdxs,
                         float* __restrict__ U, float* __restrict__ q, int V) {
    int row = blockIdx.x * blockDim.x + threadIdx.x;
    if (row >= V) return;
    const float* dr = defIdxs + (size_t)row * L;
    float* ur = U + (size_t)row * LP;
    float* qr = q + (size_t)row * LP;
    float mx = -1e30f;
    for (int j = 0; j < L; ++j) {
        float v = fminf(fmaxf(dr[j], 1e-6f), 1.0f);
        float u = __logf(v);
        ur[j] = u;
        mx = fmaxf(mx, u);
    }
    float s = 0.f;
    for (int j = 0; j < L; ++j) s += fast_exp2((ur[j] - mx) * LOG2E);
    float inv = 1.0f / s;
    for (int j = 0; j < L; ++j) qr[j] = fast_exp2((ur[j] - mx) * LOG2E) * inv;
    for (int j = L; j < LP; ++j) { qr[j] = 0.f; ur[j] = 0.f; }
}

// ---------------------------------------------------------------------------
// partial[chunk] = (FW .* gaussian) @ q over K-chunk. Diagonal is NOT zeroed
// here (branch-free loop); the exact diag term FW[r,r]*q[r,c] is subtracted in
// the reduction kernel.
// ---------------------------------------------------------------------------
__global__ void gemm_filter(const float* __restrict__ FW,
                            const float* __restrict__ P,
                            const float* __restrict__ q,
                            float* __restrict__ partial,
                            int V, int mTiles, int kSplit, int chunkLen) {
    const int lane = threadIdx.x & 31;
    const int wid  = blockIdx.x * (blockDim.x >> 5) + (threadIdx.x >> 5);
    if (wid >= mTiles * kSplit) return;          // wave-uniform: EXEC stays full
    const int mTile  = wid % mTiles;
    const int chunk  = wid / mTiles;
    const int kBegin = chunk * chunkLen;         // multiple of 4
    const int kEnd   = min(V, kBegin + chunkLen);
    const int half   = lane >> 4;                // 0 or 1
    const int l16    = lane & 15;
    const int rowU   = mTile * 16 + l16;
    const int row    = rowU < V ? rowU : V - 1;  // clamp: junk rows never stored
    const int kOff   = half * 2;
    const float px = P[(size_t)row * 3 + 0];
    const float py = P[(size_t)row * 3 + 1];
    const float pz = P[(size_t)row * 3 + 2];
    const float* fwRow = FW + (size_t)row * V;

    v8f acc0 = {}, acc1 = {}, acc2 = {};
    const int kMain = kBegin + ((kEnd - kBegin) & ~3);
    for (int k0 = kBegin; k0 < kMain; k0 += 4) {
        const int ka = k0 + kOff;
        __builtin_prefetch(fwRow + ka + 1024, 0, 0);     // ~4KB ahead on 420MB stream
        v2f fw = *(const v2f*)(fwRow + ka);              // 8B aligned (V even)
        float dx0 = px - P[(size_t)ka * 3 + 0];
        float dy0 = py - P[(size_t)ka * 3 + 1];
        float dz0 = pz - P[(size_t)ka * 3 + 2];
        float dx1 = px - P[(size_t)ka * 3 + 3];
        float dy1 = py - P[(size_t)ka * 3 + 4];
        float dz1 = pz - P[(size_t)ka * 3 + 5];
        float d0 = dx0 * dx0 + dy0 * dy0 + dz0 * dz0;
        float d1 = dx1 * dx1 + dy1 * dy1 + dz1 * dz1;
        v2f a;
        a.x = fw.x * fast_exp2(d0 * NEG2LOG2E);
        a.y = fw.y * fast_exp2(d1 * NEG2LOG2E);
        const float* q0 = q + (size_t)ka * LP + l16;
        v2f b0 = { q0[0],  q0[LP + 0]  };
        v2f b1 = { q0[16], q0[LP + 16] };
        v2f b2 = { q0[32], q0[LP + 32] };
        acc0 = WMMA_F32(a, b0, acc0);
        acc1 = WMMA_F32(a, b1, acc1);
        acc2 = WMMA_F32(a, b2, acc2);
    }
    if (kMain < kEnd) {                                  // <=3 leftover K columns
        const int ka = kMain + kOff;
        v2f a = {0.f, 0.f}, b0 = {0.f, 0.f}, b1 = {0.f, 0.f}, b2 = {0.f, 0.f};
        if (ka < kEnd) {
            float dx = px - P[(size_t)ka * 3 + 0];
            float dy = py - P[(size_t)ka * 3 + 1];
            float dz = pz - P[(size_t)ka * 3 + 2];
            a.x = fwRow[ka] * fast_exp2((dx * dx + dy * dy + dz * dz) * NEG2LOG2E);
            const float* q0 = q + (size_t)ka * LP + l16;
            b0.x = q0[0]; b1.x = q0[16]; b2.x = q0[32];
        }
        if (ka + 1 < kEnd) {
            float dx = px - P[(size_t)(ka + 1) * 3 + 0];
            float dy = py - P[(size_t)(ka + 1) * 3 + 1];
            float dz = pz - P[(size_t)(ka + 1) * 3 + 2];
            a.y = fwRow[ka + 1] * fast_exp2((dx * dx + dy * dy + dz * dz) * NEG2LOG2E);
            const float* q1 = q + (size_t)(ka + 1) * LP + l16;
            b0.y = q1[0]; b1.y = q1[16]; b2.y = q1[32];
        }
        acc0 = WMMA_F32(a, b0, acc0);
        acc1 = WMMA_F32(a, b1, acc1);
        acc2 = WMMA_F32(a, b2, acc2);
    }
    float* dst = partial + (size_t)chunk * (size_t)V * LP;
    const int orow = mTile * 16 + half * 8;
    for (int i = 0; i < 8; ++i) {
        int r = orow + i;
        if (r < V) {
            float* o = dst + (size_t)r * LP + l16;
            o[0]  = acc0[i];
            o[16] = acc1[i];
            o[32] = acc2[i];
        }
    }
}

// ---------------------------------------------------------------------------
// partial[chunk] = spatial_weights @ phi_t over K-chunk (negation in reduce).
// ---------------------------------------------------------------------------
__global__ void gemm_spatial(const float* __restrict__ SW,
                             const float* __restrict__ phiT,
                             float* __restrict__ partial,
                             int V, int mTiles, int kSplit, int chunkLen) {
    const int lane = threadIdx.x & 31;
    const int wid  = blockIdx.x * (blockDim.x >> 5) + (threadIdx.x >> 5);
    if (wid >= mTiles * kSplit) return;
    const int mTile  = wid % mTiles;
    const int chunk  = wid / mTiles;
    const int kBegin = chunk * chunkLen;
    const int kEnd   = min(V, kBegin + chunkLen);
    const int half   = lane >> 4;
    const int l16    = lane & 15;
    const int rowU   = mTile * 16 + l16;
    const int row    = rowU < V ? rowU : V - 1;
    const int kOff   = half * 2;
    const float* sRow = SW + (size_t)row * V;

    v8f acc0 = {}, acc1 = {}, acc2 = {};
    const int kMain = kBegin + ((kEnd - kBegin) & ~3);
    for (int k0 = kBegin; k0 < kMain; k0 += 4) {
        const int ka = k0 + kOff;
        __builtin_prefetch(sRow + ka + 1024, 0, 0);
        v2f a = *(const v2f*)(sRow + ka);
        const float* t0 = phiT + (size_t)ka * LP + l16;
        v2f b0 = { t0[0],  t0[LP + 0]  };
        v2f b1 = { t0[16], t0[LP + 16] };
        v2f b2 = { t0[32], t0[LP + 32] };
        acc0 = WMMA_F32(a, b0, acc0);
        acc1 = WMMA_F32(a, b1, acc1);
        acc2 = WMMA_F32(a, b2, acc2);
    }
    if (kMain < kEnd) {
        const int ka = kMain + kOff;
        v2f a = {0.f, 0.f}, b0 = {0.f, 0.f}, b1 = {0.f, 0.f}, b2 = {0.f, 0.f};
        if (ka < kEnd) {
            a.x = sRow[ka];
            const float* t0 = phiT + (size_t)ka * LP + l16;
            b0.x = t0[0]; b1.x = t0[16]; b2.x = t0[32];
        }
        if (ka + 1 < kEnd) {
            a.y = sRow[ka + 1];
            const float* t1 = phiT + (size_t)(ka + 1) * LP + l16;
            b0.y = t1[0]; b1.y = t1[16]; b2.y = t1[32];
        }
        acc0 = WMMA_F32(a, b0, acc0);
        acc1 = WMMA_F32(a, b1, acc1);
        acc2 = WMMA_F32(a, b2, acc2);
    }
    float* dst = partial + (size_t)chunk * (size_t)V * LP;
    const int orow = mTile * 16 + half * 8;
    for (int i = 0; i < 8; ++i) {
        int r = orow + i;
        if (r < V) {
            float* o = dst + (size_t)r * LP + l16;
            o[0]  = acc0[i];
            o[16] = acc1[i];
            o[32] = acc2[i];
        }
    }
}

// ---------------------------------------------------------------------------
// phiT = sum_c partial[c]  -  diag fix (filter_mat diagonal forced to 0).
// ---------------------------------------------------------------------------
__global__ void reduce_phiT(const float* __restrict__ partial,
                            const float* __restrict__ FW,
                            const float* __restrict__ q,
                            float* __restrict__ phiT, int V, int kSplit) {
    size_t n = (size_t)V * LP;
    size_t idx = (size_t)blockIdx.x * blockDim.x + threadIdx.x;
    if (idx >= n) return;
    float s = 0.f;
    for (int c = 0; c < kSplit; ++c) s += partial[(size_t)c * n + idx];
    int r = (int)(idx / LP);
    s -= FW[(size_t)r * V + r] * q[idx];   // remove k==row term (exp(0)==1)
    phiT[idx] = s;
}

// ---------------------------------------------------------------------------
// phi = -(sum_c partial[c])
// ---------------------------------------------------------------------------
__global__ void reduce_neg(const float* __restrict__ partial,
                           float* __restrict__ phi, int V, int kSplit) {
    size_t n = (size_t)V * LP;
    size_t idx = (size_t)blockIdx.x * blockDim.x + threadIdx.x;
    if (idx >= n) return;
    float s = 0.f;
    for (int c = 0; c < kSplit; ++c) s += partial[(size_t)c * n + idx];
    phi[idx] = -s;
}

// ---------------------------------------------------------------------------
// q = softmax(U - phi @ comp_matrix) ; pads stay zero. Optionally emit d_out.
// ---------------------------------------------------------------------------
__global__ void comp_softmax(const float* __restrict__ phi,
                             const float* __restrict__ C,
                             const float* __restrict__ U,
                             float* __restrict__ q,
                             float* __restrict__ out, int V) {
    int row = blockIdx.x * blockDim.x + threadIdx.x;
    if (row >= V) return;
    const float* pr = phi + (size_t)row * LP;
    const float* ur = U + (size_t)row * LP;
    float p[L];
#pragma unroll
    for (int l = 0; l < L; ++l) p[l] = pr[l];       // register-resident row

    float mx = -1e30f;
    for (int j = 0; j < L; ++j) {
        float s = 0.f;
#pragma unroll
        for (int l = 0; l < L; ++l) s += p[l] * C[l * L + j];
        mx = fmaxf(mx, ur[j] - s);
    }
    float sum = 0.f;
    for (int j = 0; j < L; ++j) {
        float s = 0.f;
#pragma unroll
        for (int l = 0; l < L; ++l) s += p[l] * C[l * L + j];
        sum += fast_exp2((ur[j] - s - mx) * LOG2E);   // arg <= 0: raw v_exp ok
    }
    float inv = 1.0f / sum;
    float* qr = q + (size_t)row * LP;
    for (int j = 0; j < L; ++j) {
        float s = 0.f;
#pragma unroll
        for (int l = 0; l < L; ++l) s += p[l] * C[l * L + j];
        float val = fast_exp2((ur[j] - s - mx) * LOG2E) * inv;
        qr[j] = val;
        if (out) out[(size_t)row * L + j] = val;
    }
}

// ---------------------------------------------------------------------------
extern "C" void kernel_launch(void* const* d_in, const int* in_sizes, int n_in,
                              void* d_out, int out_size, void* d_ws, size_t ws_size,
                              hipStream_t stream) {
    const float* defIdxs = (const float*)d_in[0];   // [V, L]
    const float* P       = (const float*)d_in[1];   // [V, 3]
    const float* FW      = (const float*)d_in[2];   // [V, V]
    const float* SW      = (const float*)d_in[3];   // [V, V]
    const float* C       = (const float*)d_in[4];   // [L, L]
    float* out = (float*)d_out;                     // [V, L]
    const int V = in_sizes[1] / 3;

    float* ws = (float*)d_ws;
    const size_t stride = (size_t)V * LP;           // floats per [V,48] buffer
    float* U       = ws;
    float* q       = ws + stride;
    float* phiT    = ws + 2 * stride;
    float* phi     = ws + 3 * stride;
    float* partial = ws + 4 * stride;

    // K-split chosen from available workspace (fixed per run => deterministic).
    const size_t availF = ws_size / sizeof(float);
    int kSplit = 1;
    if (availF > 5 * stride) {
        size_t k = (availF - 4 * stride) / stride;
        kSplit = (int)(k > 8 ? 8 : k);
        if (kSplit < 1) kSplit = 1;
    }
    const int chunkLen = ((((V + kSplit - 1) / kSplit) + 3) / 4) * 4;   // mult of 4

    const int threads = 256;                        // 8 waves / block (wave32)
    init_u_q<<<(V + threads - 1) / threads, threads, 0, stream>>>(defIdxs, U, q, V);

    const int mTiles  = (V + 15) / 16;
    const int waves   = mTiles * kSplit;
    const int wpb     = threads / 32;
    const int gBlocks = (waves + wpb - 1) / wpb;
    const int rBlocks = (int)((stride + threads - 1) / threads);

    for (int it = 0; it < NUM_ITER; ++it) {
        gemm_filter<<<gBlocks, threads, 0, stream>>>(FW, P, q, partial,
                                                     V, mTiles, kSplit, chunkLen);
        reduce_phiT<<<rBlocks, threads, 0, stream>>>(partial, FW, q, phiT, V, kSplit);
        gemm_spatial<<<gBlocks, threads, 0, stream>>>(SW, phiT, partial,
                                                      V, mTiles, kSplit, chunkLen);
        reduce_neg<<<rBlocks, threads, 0, stream>>>(partial, phi, V, kSplit);
        comp_softmax<<<(V + threads - 1) / threads, threads, 0, stream>>>(
            phi, C, U, q, (it == NUM_ITER - 1) ? out : nullptr, V);
    }
}